// SSM_60911226192464
// MI455X (gfx1250) — compile-verified
//
#include <hip/hip_runtime.h>
#include <hip/hip_bf16.h>

// ---------------- problem constants (from reference) ----------------
constexpr int BATCH  = 2;
constexpr int SEQL   = 2048;
constexpr int DIN    = 2048;   // IN_FEATURES
constexpr int DTR    = 128;    // DT_RANK
constexpr int DST    = 16;     // D_STATE
constexpr int DINNER = 2048;   // DIM_INNER
constexpr int MROWS  = BATCH * SEQL;      // 4096
constexpr int NDBC   = DTR + 2 * DST;     // 160

typedef float v2f __attribute__((ext_vector_type(2)));
typedef float v8f __attribute__((ext_vector_type(8)));

// ---------------- CDNA5 async global->LDS helpers ----------------
// GLOBAL_LOAD_ASYNC_TO_LDS_B128 (GV mode): per-lane 16B memory -> LDS,
// tracked by ASYNCcnt.
__device__ __forceinline__ void async_ld_b128(unsigned lds_byte_addr, const void* gaddr) {
  asm volatile("global_load_async_to_lds_b128 %0, %1, off"
               :: "v"(lds_byte_addr),
                  "v"((unsigned long long)(size_t)gaddr)
               : "memory");
}
__device__ __forceinline__ void s_wait_async0() {
  asm volatile("s_wait_asynccnt 0" ::: "memory");
}
__device__ __forceinline__ void s_wait_async8() {   // keep newest 8 in flight
  asm volatile("s_wait_asynccnt 8" ::: "memory");
}

// =====================================================================
// Kernel 1: dbc[m,n] = sum_k x[m,k] * W_dbc[n,k] + b_dbc[n]
//   M=4096, N=160, K=2048.  fp32 WMMA 16x16x4.
//   Wave computes a 16(M) x 80(N) strip: one A fragment feeds 5 WMMAs.
// =====================================================================
__global__ __launch_bounds__(128) void k_gemm_dbc(const float* __restrict__ x,
                                                  const float* __restrict__ Wdbc,
                                                  const float* __restrict__ bdbc,
                                                  float* __restrict__ dbc) {
  const int lane = threadIdx.x & 31;
  const int wave = threadIdx.x >> 5;
  const int m0 = blockIdx.x * 64 + wave * 16;
  const int n0 = blockIdx.y * 80;            // grid.y = 2  (160 / 80)
  const int r  = lane & 15;
  const int hi = lane >> 4;

  // A frag: x[m0+r, k0 + 2*hi + {0,1}]   (k = 2*(lane/16)+j)
  // B frag j-th tile: W_dbc[n0+16*j+r, k0 + 2*hi + {0,1}]  (B[k][n] = W[n][k])
  const float* pa = x + (size_t)(m0 + r) * DIN + 2 * hi;
  const float* pb[5];
#pragma unroll
  for (int j = 0; j < 5; ++j)
    pb[j] = Wdbc + (size_t)(n0 + 16 * j + r) * DIN + 2 * hi;

  v8f acc[5] = {};
#pragma unroll 2
  for (int k0 = 0; k0 < DIN; k0 += 4) {
    const v2f a = *(const v2f*)(pa + k0);          // one A load, 5 WMMAs
#pragma unroll
    for (int j = 0; j < 5; ++j) {
      const v2f b = *(const v2f*)(pb[j] + k0);
      acc[j] = __builtin_amdgcn_wmma_f32_16x16x4_f32(false, a, false, b,
                                                     (short)0, acc[j], false, false);
    }
  }

#pragma unroll
  for (int j = 0; j < 5; ++j) {
    const int n = n0 + 16 * j + r;
    const float bias = bdbc[n];
#pragma unroll
    for (int vr = 0; vr < 8; ++vr) {
      const int m = m0 + vr + 8 * hi;              // C/D layout
      dbc[(size_t)m * NDBC + n] = acc[j][vr] + bias;
    }
  }
}

// =====================================================================
// Kernel 2: delta[m,n] = softplus( sum_k dbc[m,k] * W_dt[n,k] + b_dt[n] )
//   M=4096, N=2048, K=128.  Wave computes 16(M) x 64(N): A frag -> 4 WMMAs.
// =====================================================================
__global__ __launch_bounds__(128) void k_gemm_dt(const float* __restrict__ dbc,
                                                 const float* __restrict__ Wdt,
                                                 const float* __restrict__ bdt,
                                                 float* __restrict__ delta) {
  const int lane = threadIdx.x & 31;
  const int wave = threadIdx.x >> 5;
  const int m0 = blockIdx.x * 64 + wave * 16;
  const int n0 = blockIdx.y * 64;            // grid.y = 32
  const int r  = lane & 15;
  const int hi = lane >> 4;

  const float* pa = dbc + (size_t)(m0 + r) * NDBC + 2 * hi;   // lda = 160
  const float* pb[4];
#pragma unroll
  for (int j = 0; j < 4; ++j)
    pb[j] = Wdt + (size_t)(n0 + 16 * j + r) * DTR + 2 * hi;   // ldb = 128

  v8f acc[4] = {};
#pragma unroll 4
  for (int k0 = 0; k0 < DTR; k0 += 4) {
    const v2f a = *(const v2f*)(pa + k0);
#pragma unroll
    for (int j = 0; j < 4; ++j) {
      const v2f b = *(const v2f*)(pb[j] + k0);
      acc[j] = __builtin_amdgcn_wmma_f32_16x16x4_f32(false, a, false, b,
                                                     (short)0, acc[j], false, false);
    }
  }

#pragma unroll
  for (int j = 0; j < 4; ++j) {
    const int n = n0 + 16 * j + r;
    const float bias = bdt[n];
#pragma unroll
    for (int vr = 0; vr < 8; ++vr) {
      const int m = m0 + vr + 8 * hi;
      const float z  = acc[j][vr] + bias;
      const float sp = fmaxf(z, 0.f) + __logf(1.f + __expf(-fabsf(z)));  // softplus
      delta[(size_t)m * DINNER + n] = sp;
    }
  }
}

// =====================================================================
// Kernel 3: fused selective scan.
//   Block = 256 thr (8 waves) handles 16 channels of one batch.
//   lane -> (channel = 2*wave + lane/16, state s = lane&15).
//   B/C/delta/x streamed via double-buffered async global->LDS chunks.
// =====================================================================
constexpr int TCH   = 128;              // timesteps per chunk
constexpr int NCHNK = SEQL / TCH;       // 16
constexpr int CHB   = 16;               // channels per block
constexpr unsigned SM_ARR = TCH * 16 * 4;  // 8192 B per array per buffer
// LDS layout (static shared assumed at LDS offset 0):
//   [0] B buf0/1, [2] C buf0/1, [4] delta buf0/1, [6] x buf0/1, [8] Y
constexpr unsigned SMO_B = 0 * SM_ARR;
constexpr unsigned SMO_C = 2 * SM_ARR;
constexpr unsigned SMO_D = 4 * SM_ARR;
constexpr unsigned SMO_X = 6 * SM_ARR;

__device__ __forceinline__ void issue_chunk_loads(int buf, int rowbase, int d0,
                                                  const float* __restrict__ dbc,
                                                  const float* __restrict__ delta,
                                                  const float* __restrict__ x,
                                                  int tid) {
  // 4 arrays x 2048 floats = 512 float4 per array; 256 threads -> 2 each.
  // 8 async instructions per wave per chunk -> ASYNCcnt += 8.
#pragma unroll
  for (int i = 0; i < 2; ++i) {
    const int f4 = tid + i * 256;           // 0..511
    const int t  = f4 >> 2;
    const int q  = (f4 & 3) * 4;            // float offset in 16-wide row
    const size_t row = (size_t)(rowbase + t);
    const unsigned e = (unsigned)f4 * 16u;  // byte offset within array
    async_ld_b128(SMO_B + (unsigned)buf * SM_ARR + e, dbc   + row * NDBC   + DTR + q);
    async_ld_b128(SMO_C + (unsigned)buf * SM_ARR + e, dbc   + row * NDBC   + DTR + DST + q);
    async_ld_b128(SMO_D + (unsigned)buf * SM_ARR + e, delta + row * DINNER + d0 + q);
    async_ld_b128(SMO_X + (unsigned)buf * SM_ARR + e, x     + row * DINNER + d0 + q);
  }
}

__global__ __launch_bounds__(256) void k_scan(const float* __restrict__ dbc,
                                              const float* __restrict__ delta,
                                              const float* __restrict__ x,
                                              const float* __restrict__ A_log,
                                              const float* __restrict__ Dvec,
                                              float* __restrict__ y) {
  __shared__ __align__(16) float smem[9 * TCH * 16];   // 72 KB of 320 KB WGP LDS
  float* const smB0 = smem;                 // matches SMO_* byte offsets
  float* const smY  = smem + 8 * TCH * 16;

  const int tid   = threadIdx.x;
  const int lane  = tid & 31;
  const int wave  = tid >> 5;
  const int s     = lane & 15;
  const int chLoc = (wave << 1) | (lane >> 4);     // 0..15
  const int b     = blockIdx.x >> 7;               // 128 blocks per batch
  const int d0    = (blockIdx.x & 127) * CHB;
  const int d     = d0 + chLoc;
  const int rowbase0 = b * SEQL;

  const float Acoef = -__expf(A_log[(size_t)d * DST + s]);  // A[d,s] = -exp(A_log)
  const float Dcoef = Dvec[d];

  issue_chunk_loads(0, rowbase0, d0, dbc, delta, x, tid);

  float h = 0.f;
  for (int c = 0; c < NCHNK; ++c) {
    const int cur = c & 1;
    if (c + 1 < NCHNK) {
      issue_chunk_loads(cur ^ 1, rowbase0 + (c + 1) * TCH, d0, dbc, delta, x, tid);
      s_wait_async8();                 // loads complete in order: chunk c ready
    } else {
      s_wait_async0();
    }
    __syncthreads();

    const float* Bc = smB0 + (2 * 0 + cur) * (TCH * 16);
    const float* Cc = smB0 + (2 * 1 + cur) * (TCH * 16);
    const float* Dl = smB0 + (2 * 2 + cur) * (TCH * 16);
    const float* Xc = smB0 + (2 * 3 + cur) * (TCH * 16);

    // serial chain is only h = fma(a, h, ...); exp inputs are h-independent,
    // so unrolling lets the compiler batch LDS reads + exps ahead of it.
#pragma unroll 4
    for (int t = 0; t < TCH; ++t) {
      const float dt = Dl[t * 16 + chLoc];   // LDS broadcast across 16 lanes
      const float xv = Xc[t * 16 + chLoc];
      const float bv = Bc[t * 16 + s];       // conflict-free
      const float cv = Cc[t * 16 + s];
      const float a  = __expf(dt * Acoef);
      h = fmaf(a, h, dt * bv * xv);          // h = exp(dA)*h + d*B*x
      float p = h * cv;                      // contract over the 16 states
      p += __shfl_xor(p, 1, 16);
      p += __shfl_xor(p, 2, 16);
      p += __shfl_xor(p, 4, 16);
      p += __shfl_xor(p, 8, 16);
      if (s == 0) smY[t * 16 + chLoc] = fmaf(Dcoef, xv, p);
    }
    __syncthreads();

    // coalesced b128 writeback of the y chunk
#pragma unroll
    for (int i = 0; i < 2; ++i) {
      const int f4 = tid + i * 256;
      const int t  = f4 >> 2;
      const int q  = (f4 & 3) * 4;
      const float4 v = *(const float4*)&smY[f4 * 4];
      *(float4*)&y[(size_t)(rowbase0 + c * TCH + t) * DINNER + d0 + q] = v;
    }
  }
}

// =====================================================================
// launch
// =====================================================================
extern "C" void kernel_launch(void* const* d_in, const int* in_sizes, int n_in,
                              void* d_out, int out_size, void* d_ws, size_t ws_size,
                              hipStream_t stream) {
  const float* x    = (const float*)d_in[0];
  const float* Wdbc = (const float*)d_in[1];
  const float* bdbc = (const float*)d_in[2];
  const float* Wdt  = (const float*)d_in[3];
  const float* bdt  = (const float*)d_in[4];
  const float* Alog = (const float*)d_in[5];
  const float* Dv   = (const float*)d_in[6];
  float* y = (float*)d_out;

  float* dbc   = (float*)d_ws;                         // 4096*160 f32 = 2.5 MB
  float* delta = dbc + (size_t)MROWS * NDBC;           // 4096*2048 f32 = 32 MB

  dim3 g1(MROWS / 64, NDBC / 80);      // 64 x 2
  k_gemm_dbc<<<g1, 128, 0, stream>>>(x, Wdbc, bdbc, dbc);

  dim3 g2(MROWS / 64, DINNER / 64);    // 64 x 32
  k_gemm_dt<<<g2, 128, 0, stream>>>(dbc, Wdt, bdt, delta);

  k_scan<<<BATCH * (DINNER / CHB), 256, 0, stream>>>(dbc, delta, x, Alog, Dv, y);
}